// CrossScalePredictor_13391708029266
// MI455X (gfx1250) — compile-verified
//
#include <hip/hip_runtime.h>
#include <cstdint>

// ---------------------------------------------------------------------------
// CrossScalePredictor scatter-add, MI455X (gfx1250, wave32).
// BATCH == 32 == wave width: work in node-major (N,32) layout so every
// per-edge gather/atomic is one coalesced 128B line.
//   ws[0 .. 12.8MB)   rho_t  (N_NODES x 32)
//   ws[12.8 .. 25.6M) out_t  (N_NODES x 32)  accumulated with f32 atomics (L2)
// ---------------------------------------------------------------------------

#define AS1 __attribute__((address_space(1)))

typedef __attribute__((ext_vector_type(4))) int v4i;

#if defined(__AMDGCN__) && __has_builtin(__builtin_amdgcn_global_load_async_to_lds_b128) && \
    __has_builtin(__builtin_amdgcn_s_wait_asynccnt)
#define USE_ASYNC 1
#else
#define USE_ASYNC 0
#endif

#if USE_ASYNC
#define WAIT_STAGED() __builtin_amdgcn_s_wait_asynccnt(0)
#else
#define WAIT_STAGED() ((void)0)
#endif

#define EDGE_TILE 1024          // edges staged per block pass
#define EPW (EDGE_TILE / 8)     // edges per wave per pass (8 waves / block)

__device__ __forceinline__ void atomic_fadd_g(float* p, float v) {
#if defined(__AMDGCN__) && __has_builtin(__builtin_amdgcn_global_atomic_fadd_f32)
  (void)__builtin_amdgcn_global_atomic_fadd_f32((AS1 float*)(void*)p, v);
#else
  unsafeAtomicAdd(p, v);   // lowers to global_atomic_add_f32 (no return)
#endif
}

// ---------------------------------------------------------------------------
// ISA probe (FIRST function in file so the disasm snippet shows the exact
// lowering of the async-to-LDS load, s_wait_asynccnt, and the f32 atomic).
// Never launched by kernel_launch; kept alive because it is __global__.
// ---------------------------------------------------------------------------
__global__ __launch_bounds__(32) void aaa_isa_probe(const int* __restrict__ g,
                                                    float* __restrict__ out,
                                                    float v) {
  __shared__ __align__(16) int buf[32 * 4];
  const int t = threadIdx.x;
#if USE_ASYNC
  __builtin_amdgcn_global_load_async_to_lds_b128((v4i*)(void*)(g + 4 * t),
                                                 (v4i*)(buf + 4 * t), 0, 0);
  WAIT_STAGED();
#else
  *(int4*)(buf + 4 * t) = *(const int4*)(g + 4 * t);
#endif
  __syncthreads();
  atomic_fadd_g(out + buf[t], v);
}

// ---------------------------------------------------------------------------
// Edge-tile staging: 256 threads stage EDGE_TILE src + dst indices into LDS.
// Full-tile check is wave-uniform (scalar branch) so the hot path carries no
// exec-mask manipulation; only the last partial tile takes the guarded path.
// ---------------------------------------------------------------------------
__device__ __forceinline__ void stage_tile(const int* __restrict__ esrc,
                                           const int* __restrict__ edst,
                                           int* sS, int* sD, int tb, int n_edges) {
  const int t = threadIdx.x;                    // 0..255, 4 edges each
  const int e0 = tb + 4 * t;
  if (tb + EDGE_TILE <= n_edges) {              // uniform: full tile
#if USE_ASYNC
    __builtin_amdgcn_global_load_async_to_lds_b128((v4i*)(void*)(esrc + e0),
                                                   (v4i*)(sS + 4 * t), 0, 0);
    __builtin_amdgcn_global_load_async_to_lds_b128((v4i*)(void*)(edst + e0),
                                                   (v4i*)(sD + 4 * t), 0, 0);
#else
    *(int4*)(sS + 4 * t) = *(const int4*)(esrc + e0);
    *(int4*)(sD + 4 * t) = *(const int4*)(edst + e0);
#endif
  } else {                                      // partial last tile
#pragma unroll
    for (int k = 0; k < 4; ++k) {
      const int e = e0 + k;
      sS[4 * t + k] = (e < n_edges) ? esrc[e] : 0;
      sD[4 * t + k] = (e < n_edges) ? edst[e] : 0;
    }
  }
}

// ---------------------------------------------------------------------------
// Hot kernel: per edge, lane b:
//   out_t[dst*32+b] += (lat ? th_lat : th_down) * rho_t[src*32+b]
//   out_t[src*32+b] += (lat ? th_lat : th_up  ) * rho_t[dst*32+b]
// Double-buffered LDS index tiles; one barrier per tile.
// ---------------------------------------------------------------------------
__global__ __launch_bounds__(256) void edge_scatter_kernel(
    const int* __restrict__ esrc, const int* __restrict__ edst,
    const float* __restrict__ rho_t, float* __restrict__ out_t,
    const float* __restrict__ p_tl, const float* __restrict__ p_tu,
    const float* __restrict__ p_td, const int* __restrict__ p_nlat,
    int n_edges, int n_tiles) {
  __shared__ __align__(16) int ldsS[2][EDGE_TILE];
  __shared__ __align__(16) int ldsD[2][EDGE_TILE];
  const float th_lat  = p_tl[0];
  const float th_up   = p_tu[0];
  const float th_down = p_td[0];
  const int   n_lat   = p_nlat[0];
  const int wave = threadIdx.x >> 5;
  const unsigned lane = threadIdx.x & 31u;
  const int stride = gridDim.x;
  const int tile0 = blockIdx.x;
  int pp = 0;
  if (tile0 < n_tiles)
    stage_tile(esrc, edst, ldsS[0], ldsD[0], tile0 * EDGE_TILE, n_edges);

  for (int t = tile0; t < n_tiles; t += stride, pp ^= 1) {
    WAIT_STAGED();                 // this wave's staged loads are in LDS
    __syncthreads();               // everyone's staged loads visible; prev tile done
    const int nxt = t + stride;
    if (nxt < n_tiles)             // overlap next tile's index fetch with compute
      stage_tile(esrc, edst, ldsS[pp ^ 1], ldsD[pp ^ 1], nxt * EDGE_TILE, n_edges);

    const int base = t * EDGE_TILE + wave * EPW;
    const int* bs = &ldsS[pp][wave * EPW];
    const int* bd = &ldsD[pp][wave * EPW];
    int cnt = n_edges - base;
    if (cnt > EPW) cnt = EPW;
#pragma unroll 8
    for (int i = 0; i < cnt; ++i) {
      const unsigned s = (unsigned)bs[i];        // uniform LDS broadcast
      const unsigned d = (unsigned)bd[i];
      const bool lat = (base + i) < n_lat;
      const float fd = lat ? th_lat : th_down;   // into dst
      const float fu = lat ? th_lat : th_up;     // into src
      const unsigned so = (s << 5) + lane;       // 32-bit offsets -> GVS addressing
      const unsigned dof = (d << 5) + lane;
      const float rs = rho_t[so];                // coalesced 128B gather
      const float rd = rho_t[dof];
      atomic_fadd_g(out_t + dof, fd * rs);
      atomic_fadd_g(out_t + so, fu * rd);
    }
  }
}

// ---------------------------------------------------------------------------
// Kernel A: rho (32 x N) -> rho_t (N x 32), and zero out_t.
// One wave transposes a 32x32 tile via a wave-private LDS tile (no barriers).
// ---------------------------------------------------------------------------
__global__ __launch_bounds__(256) void transpose_in_kernel(
    const float* __restrict__ rho, float* __restrict__ rho_t,
    float* __restrict__ out_t, int n_nodes) {
  __shared__ float tile[8][32 * 33];            // +1 pad column: no bank conflicts
  const int wave = threadIdx.x >> 5;
  const int lane = threadIdx.x & 31;
  const int n0 = (blockIdx.x * 8 + wave) * 32;
  if (n0 >= n_nodes) return;
  float* t = tile[wave];
#pragma unroll 8
  for (int b = 0; b < 32; ++b) {                // coalesced 128B reads per batch row
    const int n = n0 + lane;
    t[lane * 33 + b] = (n < n_nodes) ? rho[(size_t)b * n_nodes + n] : 0.0f;
  }
#pragma unroll 8
  for (int r = 0; r < 32; ++r) {                // coalesced 128B writes per node row
    const int n = n0 + r;
    if (n < n_nodes) {
      rho_t[((size_t)n << 5) + lane] = t[r * 33 + lane];
      out_t[((size_t)n << 5) + lane] = 0.0f;
    }
  }
}

// ---------------------------------------------------------------------------
// Kernel C: out (32 x N) = out_t (N x 32)^T + bias
// ---------------------------------------------------------------------------
__global__ __launch_bounds__(256) void transpose_out_kernel(
    const float* __restrict__ out_t, const float* __restrict__ bias,
    float* __restrict__ out, int n_nodes) {
  __shared__ float tile[8][32 * 33];
  const int wave = threadIdx.x >> 5;
  const int lane = threadIdx.x & 31;
  const int n0 = (blockIdx.x * 8 + wave) * 32;
  if (n0 >= n_nodes) return;
  const float b0 = bias[0];
  float* t = tile[wave];
#pragma unroll 8
  for (int r = 0; r < 32; ++r) {
    const int n = n0 + r;
    t[r * 33 + lane] = (n < n_nodes) ? out_t[((size_t)n << 5) + lane] : 0.0f;
  }
#pragma unroll 8
  for (int b = 0; b < 32; ++b) {
    const int n = n0 + lane;
    if (n < n_nodes) out[(size_t)b * n_nodes + n] = t[lane * 33 + b] + b0;
  }
}

// ---------------------------------------------------------------------------
extern "C" void kernel_launch(void* const* d_in, const int* in_sizes, int n_in,
                              void* d_out, int out_size, void* d_ws, size_t ws_size,
                              hipStream_t stream) {
  const float* rho     = (const float*)d_in[0];
  const float* th_lat  = (const float*)d_in[1];
  const float* th_up   = (const float*)d_in[2];
  const float* th_down = (const float*)d_in[3];
  const float* bias    = (const float*)d_in[4];
  const int*   esrc    = (const int*)d_in[5];
  const int*   edst    = (const int*)d_in[6];
  const int*   p_nlat  = (const int*)d_in[7];

  const int n_edges = in_sizes[5];
  const int n_nodes = in_sizes[0] / 32;          // BATCH == 32 (wave32)

  float* rho_t = (float*)d_ws;                   // n_nodes*32 floats
  float* out_t = rho_t + (size_t)n_nodes * 32;   // n_nodes*32 floats

  const int node_blocks = (n_nodes + 255) / 256; // 8 waves x 32 nodes / block
  transpose_in_kernel<<<node_blocks, 256, 0, stream>>>(rho, rho_t, out_t, n_nodes);

  const int n_tiles = (n_edges + EDGE_TILE - 1) / EDGE_TILE;
  int blocks = n_tiles < 2048 ? n_tiles : 2048;  // 1-2 tiles/block, double-buffered
  edge_scatter_kernel<<<blocks, 256, 0, stream>>>(esrc, edst, rho_t, out_t,
                                                  th_lat, th_up, th_down, p_nlat,
                                                  n_edges, n_tiles);

  transpose_out_kernel<<<node_blocks, 256, 0, stream>>>(out_t, bias,
                                                        (float*)d_out, n_nodes);
}